// GATWithFourier_36292473651258
// MI455X (gfx1250) — compile-verified
//
#include <hip/hip_runtime.h>
#include <hip/hip_bf16.h>
#include <math.h>

// ---------------------------------------------------------------------------
// GATWithFourier for MI455X (gfx1250, wave32).
//
// Structure:
//  K1 build_cos   : cosT[512][272] = cos(2*pi*k*f/512), zero-padded cols
//  K2 fft_gemm    : WMMA f32 16x16x4 GEMM: [4096,512] @ [512,257] for occ & prc
//                   -> xr[Nt,2]  (Nt = 4*1024*257 = 1,052,672)
//  K3 node_local  : per node: h1 = x@W1.  t>=1024: full local GAT1+ELU+GAT2
//                   (self-loop only => alpha==1) -> out2col[t].
//                   t<1024: stash h1/a_src/a_dst, init segment-max/sum bufs.
//  K4/K5          : layer-1 edge segmax (ordered-int atomicMax) + weighted sum
//  K6 n1fin       : layer-1 finalize (+self term), ELU, layer-2 per-node prep
//  K7/K8          : layer-2 edge segmax + weighted sum
//  K9 n2fin       : layer-2 finalize -> out2col[t] for t<1024
//  K10 decoder    : out[row] = sum_f out2col[row*257+f]*dec_W[f] + dec_b
// ---------------------------------------------------------------------------

#define BB   4
#define NN   1024
#define SEQ  512
#define FOUR 257
#define FOURP 272              // padded cols (17 tiles of 16)
#define ROWS (BB*NN)           // 4096 GEMM rows
#define NT_TOTAL (ROWS*FOUR)   // 1,052,672 graph nodes
#define EDGES 32768
#define H1 3
#define C1 10
#define F1 (H1*C1)             // 30
#define NEG_SLOPE 0.2f

typedef __attribute__((ext_vector_type(2))) float v2f;
typedef __attribute__((ext_vector_type(8))) float v8f;

__device__ __forceinline__ int f2o(float f) {      // monotone float->int
    int i = __float_as_int(f);
    return i >= 0 ? i : (i ^ 0x7fffffff);
}
__device__ __forceinline__ float o2f(int i) {      // involution
    return __int_as_float(i >= 0 ? i : (i ^ 0x7fffffff));
}
__device__ __forceinline__ float leaky(float x) { return x > 0.f ? x : NEG_SLOPE * x; }
__device__ __forceinline__ float elu(float x)   { return x > 0.f ? x : (expf(x) - 1.f); }

// ---------------- K1: cosine table --------------------------------------
__global__ __launch_bounds__(256) void build_cos(float* __restrict__ cosT) {
    int idx = blockIdx.x * 256 + threadIdx.x;         // k*FOURP + f
    if (idx >= SEQ * FOURP) return;
    int k = idx / FOURP, f = idx % FOURP;
    float v = 0.f;
    if (f < FOUR) {
        int ph = (k * f) & (SEQ - 1);                 // exact integer phase mod 512
        v = cosf((float)ph * 0.012271846303085130f);  // 2*pi/512
    }
    cosT[idx] = v;
}

// ---------------- K2: WMMA fp32 GEMM (rfft real part) --------------------
// One wave computes a 16x16 tile of BOTH occ- and prc- outputs (shared B).
__global__ __launch_bounds__(256) void fft_gemm(const float* __restrict__ occ,
                                                const float* __restrict__ prc,
                                                const float* __restrict__ cosT,
                                                float* __restrict__ xr) {
    const int lane = threadIdx.x & 31;
    const int waveId = blockIdx.x * 8 + (threadIdx.x >> 5);
    const int NTILES = FOURP / 16;                    // 17
    const int mt = waveId / NTILES;
    const int nt = waveId % NTILES;
    if (mt >= ROWS / 16) return;                      // whole-wave uniform exit
    const int half = lane >> 4;                       // 0 | 1
    const int l    = lane & 15;
    const int row  = mt * 16 + l;                     // A row for this lane
    const int col  = nt * 16 + l;                     // B/C column for this lane
    const float* aO = occ + (size_t)row * SEQ;
    const float* aP = prc + (size_t)row * SEQ;

    v8f accO = {}; v8f accP = {};
    for (int k0 = 0; k0 < SEQ; k0 += 4) {
        const int ka = k0 + half * 2;                 // lanes 16-31 hold K+2,K+3
        v2f fa; fa.x = aO[ka];  fa.y = aO[ka + 1];
        v2f fp; fp.x = aP[ka];  fp.y = aP[ka + 1];
        v2f fb; fb.x = cosT[(size_t)ka * FOURP + col];
                fb.y = cosT[(size_t)(ka + 1) * FOURP + col];
        accO = __builtin_amdgcn_wmma_f32_16x16x4_f32(false, fa, false, fb,
                                                     (short)0, accO, false, false);
        accP = __builtin_amdgcn_wmma_f32_16x16x4_f32(false, fp, false, fb,
                                                     (short)0, accP, false, false);
    }
    // C/D layout: VGPR r -> row (r + half*8), col = l
    const int f = nt * 16 + l;
    if (f < FOUR) {
#pragma unroll
        for (int r = 0; r < 8; ++r) {
            int m = mt * 16 + r + half * 8;
            size_t base = ((size_t)m * FOUR + f) * 2;
            xr[base + 0] = accO[r];
            xr[base + 1] = accP[r];
        }
    }
}

// ---------------- K3: per-node local pipeline / hot-node prep ------------
__global__ __launch_bounds__(256) void node_local(
    const float* __restrict__ xr,  const float* __restrict__ W1,
    const float* __restrict__ b1,  const float* __restrict__ aS1,
    const float* __restrict__ aD1, const float* __restrict__ W2,
    const float* __restrict__ b2,
    float* __restrict__ out2col, float* __restrict__ h1s,
    float* __restrict__ asrc1,   float* __restrict__ adst1,
    float* __restrict__ eself1,  int* __restrict__ m1,
    float* __restrict__ den1,    float* __restrict__ num1) {
    int t = blockIdx.x * 256 + threadIdx.x;
    if (t >= NT_TOTAL) return;
    float x0 = xr[(size_t)t * 2], x1 = xr[(size_t)t * 2 + 1];
    float h[F1];
#pragma unroll
    for (int j = 0; j < F1; ++j) h[j] = x0 * W1[j] + x1 * W1[F1 + j];
    if (t >= NN) {
        // self-loop only: softmax over a single edge -> alpha = 1 both layers
        float acc = 0.f;
#pragma unroll
        for (int j = 0; j < F1; ++j) acc += elu(h[j] + b1[j]) * W2[j];
        out2col[t] = acc + b2[0];
    } else {
#pragma unroll
        for (int j = 0; j < F1; ++j) { h1s[t * F1 + j] = h[j]; num1[t * F1 + j] = 0.f; }
#pragma unroll
        for (int hh = 0; hh < H1; ++hh) {
            float as = 0.f, ad = 0.f;
#pragma unroll
            for (int c = 0; c < C1; ++c) {
                as += h[hh * C1 + c] * aS1[hh * C1 + c];
                ad += h[hh * C1 + c] * aD1[hh * C1 + c];
            }
            asrc1[t * H1 + hh] = as; adst1[t * H1 + hh] = ad;
            float e = leaky(as + ad);                 // self-loop logit
            eself1[t * H1 + hh] = e;
            m1[t * H1 + hh] = f2o(e);                 // init segment max
            den1[t * H1 + hh] = 0.f;
        }
    }
}

// ---------------- K4/K5: layer-1 edge passes -----------------------------
__global__ __launch_bounds__(256) void e1max(const int* __restrict__ ei,
                                             const float* __restrict__ asrc1,
                                             const float* __restrict__ adst1,
                                             int* __restrict__ m1) {
    int e = blockIdx.x * 256 + threadIdx.x;
    if (e >= EDGES) return;
    int s = ei[e], d = ei[EDGES + e];
#pragma unroll
    for (int h = 0; h < H1; ++h) {
        float v = leaky(asrc1[s * H1 + h] + adst1[d * H1 + h]);
        atomicMax(&m1[d * H1 + h], f2o(v));
    }
}

__global__ __launch_bounds__(256) void e1sum(const int* __restrict__ ei,
                                             const float* __restrict__ asrc1,
                                             const float* __restrict__ adst1,
                                             const int* __restrict__ m1,
                                             const float* __restrict__ h1s,
                                             float* __restrict__ den1,
                                             float* __restrict__ num1) {
    int e = blockIdx.x * 256 + threadIdx.x;
    if (e >= EDGES) return;
    int s = ei[e], d = ei[EDGES + e];
#pragma unroll
    for (int h = 0; h < H1; ++h) {
        float v = leaky(asrc1[s * H1 + h] + adst1[d * H1 + h]);
        float w = expf(v - o2f(m1[d * H1 + h]));
        atomicAdd(&den1[d * H1 + h], w);
#pragma unroll
        for (int c = 0; c < C1; ++c)
            atomicAdd(&num1[d * F1 + h * C1 + c], h1s[s * F1 + h * C1 + c] * w);
    }
}

// ---------------- K6: layer-1 finalize + layer-2 prep (t < 1024) ---------
__global__ __launch_bounds__(256) void n1fin(
    const float* __restrict__ h1s,   const float* __restrict__ eself1,
    const int* __restrict__ m1,      const float* __restrict__ den1,
    const float* __restrict__ num1,  const float* __restrict__ b1,
    const float* __restrict__ W2,    const float* __restrict__ aS2,
    const float* __restrict__ aD2,
    float* __restrict__ y1s, float* __restrict__ h2s,
    float* __restrict__ asrc2, float* __restrict__ adst2,
    float* __restrict__ eself2, int* __restrict__ m2,
    float* __restrict__ den2, float* __restrict__ num2) {
    int t = blockIdx.x * 256 + threadIdx.x;
    if (t >= NN) return;
    float y[F1];
#pragma unroll
    for (int hh = 0; hh < H1; ++hh) {
        float ws = expf(eself1[t * H1 + hh] - o2f(m1[t * H1 + hh]));
        float dn = den1[t * H1 + hh] + ws;
#pragma unroll
        for (int c = 0; c < C1; ++c) {
            int j = hh * C1 + c;
            float o = (num1[t * F1 + j] + h1s[t * F1 + j] * ws) / dn + b1[j];
            y[j] = elu(o);
        }
    }
    float h2 = 0.f;
#pragma unroll
    for (int j = 0; j < F1; ++j) { y1s[t * F1 + j] = y[j]; h2 += y[j] * W2[j]; }
    h2s[t] = h2;
    float as = h2 * aS2[0], ad = h2 * aD2[0];
    asrc2[t] = as; adst2[t] = ad;
    float e = leaky(as + ad);
    eself2[t] = e; m2[t] = f2o(e); den2[t] = 0.f; num2[t] = 0.f;
}

// ---------------- K7/K8: layer-2 edge passes -----------------------------
__global__ __launch_bounds__(256) void e2max(const int* __restrict__ ei,
                                             const float* __restrict__ asrc2,
                                             const float* __restrict__ adst2,
                                             int* __restrict__ m2) {
    int e = blockIdx.x * 256 + threadIdx.x;
    if (e >= EDGES) return;
    int s = ei[e], d = ei[EDGES + e];
    atomicMax(&m2[d], f2o(leaky(asrc2[s] + adst2[d])));
}

__global__ __launch_bounds__(256) void e2sum(const int* __restrict__ ei,
                                             const float* __restrict__ asrc2,
                                             const float* __restrict__ adst2,
                                             const int* __restrict__ m2,
                                             const float* __restrict__ h2s,
                                             float* __restrict__ den2,
                                             float* __restrict__ num2) {
    int e = blockIdx.x * 256 + threadIdx.x;
    if (e >= EDGES) return;
    int s = ei[e], d = ei[EDGES + e];
    float w = expf(leaky(asrc2[s] + adst2[d]) - o2f(m2[d]));
    atomicAdd(&den2[d], w);
    atomicAdd(&num2[d], h2s[s] * w);
}

// ---------------- K9: layer-2 finalize (t < 1024) ------------------------
__global__ __launch_bounds__(256) void n2fin(const float* __restrict__ eself2,
                                             const int* __restrict__ m2,
                                             const float* __restrict__ den2,
                                             const float* __restrict__ num2,
                                             const float* __restrict__ h2s,
                                             const float* __restrict__ b2,
                                             float* __restrict__ out2col) {
    int t = blockIdx.x * 256 + threadIdx.x;
    if (t >= NN) return;
    float ws = expf(eself2[t] - o2f(m2[t]));
    out2col[t] = (num2[t] + h2s[t] * ws) / (den2[t] + ws) + b2[0];
}

// ---------------- K10: decoder -------------------------------------------
__global__ __launch_bounds__(256) void decoder(const float* __restrict__ out2col,
                                               const float* __restrict__ decW,
                                               const float* __restrict__ decb,
                                               float* __restrict__ out) {
    __shared__ float red[256];
    int row = blockIdx.x, tid = threadIdx.x;
    const float* p = out2col + (size_t)row * FOUR;
    float s = 0.f;
    for (int f = tid; f < FOUR; f += 256) s += p[f] * decW[f];
    red[tid] = s; __syncthreads();
    for (int off = 128; off > 0; off >>= 1) {
        if (tid < off) red[tid] += red[tid + off];
        __syncthreads();
    }
    if (tid == 0) out[row] = red[0] + decb[0];
}

// ---------------- host launcher ------------------------------------------
extern "C" void kernel_launch(void* const* d_in, const int* in_sizes, int n_in,
                              void* d_out, int out_size, void* d_ws, size_t ws_size,
                              hipStream_t stream) {
    const float* occ  = (const float*)d_in[0];
    const float* prc  = (const float*)d_in[1];
    const int*   ei   = (const int*)  d_in[2];
    const float* W1   = (const float*)d_in[3];
    const float* aS1  = (const float*)d_in[4];
    const float* aD1  = (const float*)d_in[5];
    const float* b1   = (const float*)d_in[6];
    const float* W2   = (const float*)d_in[7];
    const float* aS2  = (const float*)d_in[8];
    const float* aD2  = (const float*)d_in[9];
    const float* b2   = (const float*)d_in[10];
    const float* decW = (const float*)d_in[11];
    const float* decb = (const float*)d_in[12];
    float* out = (float*)d_out;

    float* ws = (float*)d_ws;
    size_t off = 0;
    float* cosT    = ws + off; off += (size_t)SEQ * FOURP;       // 139,264
    float* xr      = ws + off; off += (size_t)NT_TOTAL * 2;      // 2,105,344
    float* out2col = ws + off; off += NT_TOTAL;                  // 1,052,672
    float* h1s     = ws + off; off += NN * F1;
    float* asrc1   = ws + off; off += NN * H1;
    float* adst1   = ws + off; off += NN * H1;
    float* eself1  = ws + off; off += NN * H1;
    int*   m1      = (int*)(ws + off); off += NN * H1;
    float* den1    = ws + off; off += NN * H1;
    float* num1    = ws + off; off += NN * F1;
    float* y1s     = ws + off; off += NN * F1;
    float* h2s     = ws + off; off += NN;
    float* asrc2   = ws + off; off += NN;
    float* adst2   = ws + off; off += NN;
    float* eself2  = ws + off; off += NN;
    int*   m2      = (int*)(ws + off); off += NN;
    float* den2    = ws + off; off += NN;
    float* num2    = ws + off; off += NN;                        // ~13.6 MB total

    build_cos<<<(SEQ * FOURP + 255) / 256, 256, 0, stream>>>(cosT);

    // 256 M-tiles x 17 N-tiles = 4352 waves, 8 waves/block -> 544 blocks
    fft_gemm<<<544, 256, 0, stream>>>(occ, prc, cosT, xr);

    node_local<<<(NT_TOTAL + 255) / 256, 256, 0, stream>>>(
        xr, W1, b1, aS1, aD1, W2, b2,
        out2col, h1s, asrc1, adst1, eself1, m1, den1, num1);

    e1max<<<EDGES / 256, 256, 0, stream>>>(ei, asrc1, adst1, m1);
    e1sum<<<EDGES / 256, 256, 0, stream>>>(ei, asrc1, adst1, m1, h1s, den1, num1);
    n1fin<<<NN / 256, 256, 0, stream>>>(h1s, eself1, m1, den1, num1, b1, W2,
                                        aS2, aD2, y1s, h2s, asrc2, adst2,
                                        eself2, m2, den2, num2);
    e2max<<<EDGES / 256, 256, 0, stream>>>(ei, asrc2, adst2, m2);
    e2sum<<<EDGES / 256, 256, 0, stream>>>(ei, asrc2, adst2, m2, h2s, den2, num2);
    n2fin<<<NN / 256, 256, 0, stream>>>(eself2, m2, den2, num2, h2s, b2, out2col);

    decoder<<<ROWS, 256, 0, stream>>>(out2col, decW, decb, out);
}